// RiskGCN_55731495633543
// MI455X (gfx1250) — compile-verified
//
#include <hip/hip_runtime.h>

typedef __attribute__((ext_vector_type(2))) float v2f;
typedef __attribute__((ext_vector_type(8))) float v8f;

#define N_NODES 100000
#define WAVES_PER_BLOCK 8

// ---------------------------------------------------------------------------
// K0: init scratch. deg=1 (self loop), agg1=0, agg2=0.
__global__ void k_init(unsigned* __restrict__ deg, float* __restrict__ agg1,
                       float* __restrict__ agg2, int n) {
    int i = blockIdx.x * blockDim.x + threadIdx.x;
    if (i < n)       deg[i]  = 1u;
    if (i < n * 16)  agg1[i] = 0.0f;
    if (i < n * 2)   agg2[i] = 0.0f;
}

// K1: degree count over real edges (dst half of edge_index).
__global__ void k_deg(const long long* __restrict__ dst, unsigned* __restrict__ deg, int E) {
    int e = blockIdx.x * blockDim.x + threadIdx.x;
    if (e < E) atomicAdd(&deg[(int)dst[e]], 1u);
}

// K2: dinv = rsqrt(deg), in place (uint -> float reinterpret of same buffer).
__global__ void k_dinv(unsigned* __restrict__ degU, float* __restrict__ dinvF, int n) {
    int i = blockIdx.x * blockDim.x + threadIdx.x;
    if (i < n) {
        float d = (float)degU[i];
        dinvF[i] = rsqrtf(d);   // deg >= 1 always (self loop), matches where(deg>0)
    }
}

// ---------------------------------------------------------------------------
// K3: h1 = x @ W1 via V_WMMA_F32_16X16X4_F32. One wave per 16-node tile.
//  A (16x4 f32, 2 VGPRs): lane L: m=L%16, half=L/16; a.x = X[m][2*half], a.y = X[m][2*half+1]
//  B (4x16 f32, 2 VGPRs): lane L: b.x = W1[2*half][m], b.y = W1[2*half+1][m]
//  D (16x16 f32, 8 VGPRs): d[j] = D[j + 8*half][m]
__global__ __launch_bounds__(256) void k_xform1(const float* __restrict__ x,
                                                const float* __restrict__ W1,
                                                float* __restrict__ h1, int nTiles) {
    int wave = blockIdx.x * WAVES_PER_BLOCK + (threadIdx.x >> 5);
    if (wave >= nTiles) return;                 // wave-uniform: EXEC all-1s at WMMA
    int lane = threadIdx.x & 31;
    int half = lane >> 4;
    int m    = lane & 15;
    int row  = wave * 16 + m;

    v2f a, b;
    a.x = x[row * 4 + 2 * half + 0];
    a.y = x[row * 4 + 2 * half + 1];
    b.x = W1[(2 * half + 0) * 16 + m];
    b.y = W1[(2 * half + 1) * 16 + m];

    v8f c = {};
    c = __builtin_amdgcn_wmma_f32_16x16x4_f32(false, a, false, b, (short)0, c, false, false);

    int base = wave * 16;
#pragma unroll
    for (int j = 0; j < 8; ++j) {
        int M = j + half * 8;
        h1[(size_t)(base + M) * 16 + m] = c[j];
    }
}

// K4: g1 = h1 * dinv[node]  (in place). Also used for g2 with stride 2.
__global__ void k_scale(float* __restrict__ g, const float* __restrict__ dinv,
                        int total, int log2stride) {
    int i = blockIdx.x * blockDim.x + threadIdx.x;
    if (i < total) g[i] *= dinv[i >> log2stride];
}

// K5: edge scatter, 16 features: agg1[dst] += g1[src].
__global__ void k_scatter16(const long long* __restrict__ src,
                            const long long* __restrict__ dst,
                            const float* __restrict__ g1,
                            float* __restrict__ agg1, int E) {
    int e = blockIdx.x * blockDim.x + threadIdx.x;
    if (e >= E) return;
    int s = (int)src[e];
    int d = (int)dst[e];
    const float4* gs = (const float4*)(g1 + (size_t)s * 16);
    float* ad = agg1 + (size_t)d * 16;
#pragma unroll
    for (int q = 0; q < 4; ++q) {
        float4 v = gs[q];
        unsafeAtomicAdd(ad + 4 * q + 0, v.x);
        unsafeAtomicAdd(ad + 4 * q + 1, v.y);
        unsafeAtomicAdd(ad + 4 * q + 2, v.z);
        unsafeAtomicAdd(ad + 4 * q + 3, v.w);
    }
}

// K6: h2 = relu(dinv*(agg1 + g1) + b1), written in place over agg1.
__global__ void k_combine1(float* __restrict__ agg1, const float* __restrict__ g1,
                           const float* __restrict__ dinv, const float* __restrict__ b1,
                           int total) {
    int i = blockIdx.x * blockDim.x + threadIdx.x;
    if (i >= total) return;
    int node = i >> 4;
    float v = dinv[node] * (agg1[i] + g1[i]) + b1[i & 15];
    agg1[i] = fmaxf(v, 0.0f);
}

// K7: z = h2 @ W2 (16x2, zero-padded to 16x16), K=16 as 4 accumulating K=4 WMMAs.
__global__ __launch_bounds__(256) void k_xform2(const float* __restrict__ h2,
                                                const float* __restrict__ W2,
                                                float* __restrict__ z, int nTiles) {
    int wave = blockIdx.x * WAVES_PER_BLOCK + (threadIdx.x >> 5);
    if (wave >= nTiles) return;
    int lane = threadIdx.x & 31;
    int half = lane >> 4;
    int m    = lane & 15;
    int row  = wave * 16 + m;

    v8f c = {};
#pragma unroll
    for (int q = 0; q < 4; ++q) {
        int k0 = 4 * q + 2 * half;
        v2f a, b;
        a.x = h2[(size_t)row * 16 + k0 + 0];
        a.y = h2[(size_t)row * 16 + k0 + 1];
        // W2 is 16x2; pad columns 2..15 with zero
        b.x = (m < 2) ? W2[(k0 + 0) * 2 + m] : 0.0f;
        b.y = (m < 2) ? W2[(k0 + 1) * 2 + m] : 0.0f;
        c = __builtin_amdgcn_wmma_f32_16x16x4_f32(false, a, false, b, (short)0, c, false, false);
    }

    if (m < 2) {
#pragma unroll
        for (int j = 0; j < 8; ++j) {
            int M = j + half * 8;
            z[(size_t)(wave * 16 + M) * 2 + m] = c[j];
        }
    }
}

// K9: edge scatter, 2 features: agg2[dst] += g2[src].
__global__ void k_scatter2(const long long* __restrict__ src,
                           const long long* __restrict__ dst,
                           const float* __restrict__ g2,
                           float* __restrict__ agg2, int E) {
    int e = blockIdx.x * blockDim.x + threadIdx.x;
    if (e >= E) return;
    int s = (int)src[e];
    int d = (int)dst[e];
    float2 v = *(const float2*)(g2 + (size_t)s * 2);
    float* ad = agg2 + (size_t)d * 2;
    unsafeAtomicAdd(ad + 0, v.x);
    unsafeAtomicAdd(ad + 1, v.y);
}

// K10: out = dinv*(agg2 + g2) + b2
__global__ void k_final(const float* __restrict__ agg2, const float* __restrict__ g2,
                        const float* __restrict__ dinv, const float* __restrict__ b2,
                        float* __restrict__ out, int total) {
    int i = blockIdx.x * blockDim.x + threadIdx.x;
    if (i >= total) return;
    out[i] = dinv[i >> 1] * (agg2[i] + g2[i]) + b2[i & 1];
}

// ---------------------------------------------------------------------------
extern "C" void kernel_launch(void* const* d_in, const int* in_sizes, int n_in,
                              void* d_out, int out_size, void* d_ws, size_t ws_size,
                              hipStream_t stream) {
    const float*     x    = (const float*)d_in[0];
    const long long* eidx = (const long long*)d_in[1];   // [2, E] int64
    const float*     W1   = (const float*)d_in[2];       // [4,16]
    const float*     b1   = (const float*)d_in[3];       // [16]
    const float*     W2   = (const float*)d_in[4];       // [16,2]
    const float*     b2   = (const float*)d_in[5];       // [2]

    const int N = in_sizes[0] / 4;        // 100000 (divisible by 16)
    const int E = in_sizes[1] / 2;        // 6400000
    const long long* src = eidx;
    const long long* dst = eidx + E;

    // workspace layout (floats): dinv[N] | g1[16N] | agg1[16N] | g2[2N] | agg2[2N]
    float* ws    = (float*)d_ws;
    float* dinv  = ws;
    float* g1    = ws + (size_t)N;        // holds h1, then scaled in place
    float* agg1  = ws + (size_t)17 * N;   // holds agg, then h2 in place
    float* g2    = ws + (size_t)33 * N;   // holds z, then scaled in place
    float* agg2  = ws + (size_t)35 * N;
    unsigned* degU = (unsigned*)dinv;

    const int T = 256;
    const int nTiles     = N / 16;                                  // 6250
    const int gridTiles  = (nTiles + WAVES_PER_BLOCK - 1) / WAVES_PER_BLOCK;

    // degrees + normalization
    k_init<<<(N * 16 + T - 1) / T, T, 0, stream>>>(degU, agg1, agg2, N);
    k_deg<<<(E + T - 1) / T, T, 0, stream>>>(dst, degU, E);
    k_dinv<<<(N + T - 1) / T, T, 0, stream>>>(degU, dinv, N);

    // layer 1: h1 = x@W1 (WMMA), g1 = h1*dinv, scatter, combine+relu -> h2
    k_xform1<<<gridTiles, T, 0, stream>>>(x, W1, g1, nTiles);
    k_scale<<<(N * 16 + T - 1) / T, T, 0, stream>>>(g1, dinv, N * 16, 4);
    k_scatter16<<<(E + T - 1) / T, T, 0, stream>>>(src, dst, g1, agg1, E);
    k_combine1<<<(N * 16 + T - 1) / T, T, 0, stream>>>(agg1, g1, dinv, b1, N * 16);

    // layer 2: z = h2@W2 (WMMA), g2 = z*dinv, scatter, final combine
    k_xform2<<<gridTiles, T, 0, stream>>>(agg1, W2, g2, nTiles);
    k_scale<<<(N * 2 + T - 1) / T, T, 0, stream>>>(g2, dinv, N * 2, 1);
    k_scatter2<<<(E + T - 1) / T, T, 0, stream>>>(src, dst, g2, agg2, E);
    k_final<<<(N * 2 + T - 1) / T, T, 0, stream>>>(agg2, g2, dinv, b2, (float*)d_out, N * 2);
}